// CgpHmmCell_onedim_62088047231232
// MI455X (gfx1250) — compile-verified
//
#include <hip/hip_runtime.h>
#include <math.h>

// ---------------------------------------------------------------------------
// HMM forward (CgpHmmCell, nCodons=2): batch=2048, T=2000, 24 states.
// Wave32 design: one wave per 16 batch rows; per-step 16x24 @ 24x32 matvec
// done with 12 chained V_WMMA_F32_16X16X4_F32; alpha transposed through LDS;
// emission gathers + ds_bpermute context broadcast + shfl_xor normalization.
// ---------------------------------------------------------------------------

#define NSTATES 24
#define SPAD    32          // padded state count (WMMA N/K tiles)
#define NEMIT   17
#define BATCH   2048
#define WPB     4           // waves per block
#define AROW    34          // alpha LDS row stride (dwords), bank-conflict-free
#define BROW    65          // emission LDS row stride (dwords), conflict-free

typedef float v2f __attribute__((ext_vector_type(2)));
typedef float v8f __attribute__((ext_vector_type(8)));

// workspace layout (floats):
//   [0    , 1024)  Apad : 32x32 row-major padded transition matrix
//   [1024 , 3072)  Btab : 32x64 emission table (rows 17..23 = 1/6, 24..31 = 0)
//   [3072 , 3104)  Pi   : 32 padded initial distribution

__global__ void prep_kernel(const float* __restrict__ w,
                            const float* __restrict__ ew,
                            const float* __restrict__ ik,
                            float* __restrict__ ws)
{
    float* Apad = ws;
    float* Btab = ws + 1024;
    float* Pi   = ws + 3072;
    __shared__ float M[NSTATES][NSTATES];
    const int tid = threadIdx.x;

    for (int i = tid; i < NSTATES * NSTATES; i += blockDim.x)
        M[i / NSTATES][i % NSTATES] = -1e30f;
    __syncthreads();

    if (tid == 0) {
        // structural nonzeros for nCodons = 2 (faithful index/value port)
        M[0][0]  = 1.f - w[0];  M[0][1]  = w[0];
        M[1][2]  = 1.f;         M[2][3]  = 1.f;
        M[3][4]  = w[1];        M[6][7]  = w[2];
        M[4][5]  = 1.f;         M[7][8]  = 1.f;
        M[5][6]  = 1.f;         M[8][9]  = 1.f;
        M[3][14] = w[3];        M[6][17] = w[4];       M[9][20] = w[5];
        M[9][10] = 1.f - w[5];
        M[14][15] = 1.f;        M[17][18] = 1.f;       M[20][21] = 1.f;
        M[15][16] = 1.f;        M[18][19] = 1.f;       M[21][22] = 1.f;
        M[16][4] = w[6];        M[19][7] = w[7];       M[22][10] = w[8];
        M[16][14] = 1.f - w[9]; M[19][17] = 1.f - w[10]; M[22][20] = 1.f - w[11];
        const float wd = w[12];
        M[3][7]  = 1.f - wd * wd;
        M[3][10] = 1.f - wd * wd * wd;
        M[6][10] = 1.f - wd * wd;
        M[10][11] = 1.f; M[11][12] = 1.f; M[12][13] = 1.f;
        M[13][13] = 1.f; M[13][23] = 1.f;
        M[23][23] = 1.f;
    }
    __syncthreads();

    // row softmax of M -> Apad (exp(-1e30 - max) underflows to exact 0)
    if (tid < SPAD) {
        if (tid < NSTATES) {
            float mx = -1e30f;
            for (int j = 0; j < NSTATES; ++j) mx = fmaxf(mx, M[tid][j]);
            float e[NSTATES]; float s = 0.f;
            for (int j = 0; j < NSTATES; ++j) { e[j] = expf(M[tid][j] - mx); s += e[j]; }
            const float rs = 1.f / s;
            for (int j = 0; j < NSTATES; ++j) Apad[tid * SPAD + j] = e[j] * rs;
            for (int j = NSTATES; j < SPAD; ++j) Apad[tid * SPAD + j] = 0.f;
        } else {
            for (int j = 0; j < SPAD; ++j) Apad[tid * SPAD + j] = 0.f;
        }
    }

    // emission softmax: 17 states x 16 (pp,p) groups, softmax over 4 symbols
    for (int g = tid; g < NEMIT * 16; g += blockDim.x) {
        const int state = g >> 4, pc = g & 15;
        const float* src = ew + g * 4;
        const float mx = fmaxf(fmaxf(src[0], src[1]), fmaxf(src[2], src[3]));
        const float e0 = expf(src[0] - mx), e1 = expf(src[1] - mx);
        const float e2 = expf(src[2] - mx), e3 = expf(src[3] - mx);
        const float rs = 1.f / (e0 + e1 + e2 + e3);
        float* dst = Btab + state * 64 + pc * 4;
        dst[0] = e0 * rs; dst[1] = e1 * rs; dst[2] = e2 * rs; dst[3] = e3 * rs;
    }
    // non-emitting states 17..23 -> 1/6 ; padding states 24..31 -> 0
    for (int idx = tid + NEMIT * 64; idx < SPAD * 64; idx += blockDim.x)
        Btab[idx] = (idx < NSTATES * 64) ? (1.f / 6.f) : 0.f;

    if (tid == 0) {
        float mx = -1e30f;
        for (int j = 0; j < NSTATES; ++j) mx = fmaxf(mx, ik[j]);
        float e[NSTATES]; float s = 0.f;
        for (int j = 0; j < NSTATES; ++j) { e[j] = expf(ik[j] - mx); s += e[j]; }
        const float rs = 1.f / s;
        for (int j = 0; j < NSTATES; ++j) Pi[j] = e[j] * rs;
        for (int j = NSTATES; j < SPAD; ++j) Pi[j] = 0.f;
    }
}

__global__ void __launch_bounds__(WPB * 32, 1)
hmm_forward_kernel(const int* __restrict__ seq,
                   const float* __restrict__ ws,
                   float* __restrict__ out, int T)
{
    const float* Apad = ws;
    const float* Pi   = ws + 3072;

    __shared__ float btabS[SPAD * BROW];
    __shared__ float alphaS[WPB][16 * AROW];

    const int tid  = threadIdx.x;
    const int lane = tid & 31;
    const int wv   = tid >> 5;
    const int hl   = lane >> 4;      // wave half: 0 or 1
    const int lm   = lane & 15;

    // stage emission table into LDS with 65-dword rows (conflict-free gathers)
    for (int i = tid; i < SPAD * 64; i += blockDim.x)
        btabS[(i >> 6) * BROW + (i & 63)] = ws[1024 + i];
    __syncthreads();

    const int batchbase = (blockIdx.x * WPB + wv) * 16;

    // stationary B-operand fragments of the transition matrix:
    // bf[k][n] holds Apad[K..K+1][16n + lm], K = 4k + 2*hl
    v2f bf[6][2];
    #pragma unroll
    for (int k = 0; k < 6; ++k) {
        #pragma unroll
        for (int n = 0; n < 2; ++n) {
            const int K = 4 * k + 2 * hl;
            const int N = 16 * n + lm;
            v2f t;
            t.x = Apad[K * SPAD + N];
            t.y = Apad[(K + 1) * SPAD + N];
            bf[k][n] = t;
        }
    }

    float* aw = &alphaS[wv][0];

    // alpha_0 = pi for all 16 rows of this wave
    const float pv = Pi[lane];
    #pragma unroll 4
    for (int b = 0; b < 16; ++b) aw[b * AROW + lane] = pv;
    asm volatile("s_wait_dscnt 0x0" ::: "memory");

    // pred = alpha @ A : 12 chained f32 WMMAs (6 K-chunks x 2 N-tiles)
    auto matvec = [&](v8f& d0, v8f& d1) {
        v2f af[6];
        #pragma unroll
        for (int k = 0; k < 6; ++k)
            af[k] = *(const v2f*)(aw + lm * AROW + 4 * k + 2 * hl);
        #pragma unroll
        for (int k = 0; k < 6; ++k) {
            d0 = __builtin_amdgcn_wmma_f32_16x16x4_f32(false, af[k], false, bf[k][0],
                                                       (short)0, d0, false, false);
            d1 = __builtin_amdgcn_wmma_f32_16x16x4_f32(false, af[k], false, bf[k][1],
                                                       (short)0, d1, false, false);
        }
    };

    // t = 0,1 collapse exactly: E is uniform 1/6 (start context), rows of A and
    // alpha sum to 1 => z = 1/6 twice and alpha_1 = pi @ A.
    {
        v8f d0 = {}; v8f d1 = {};
        matvec(d0, d1);
        #pragma unroll
        for (int r = 0; r < 8; ++r) {
            const int m = r + 8 * hl;
            aw[m * AROW + lm]      = d0[r];
            aw[m * AROW + 16 + lm] = d1[r];
        }
        asm volatile("s_wait_dscnt 0x0" ::: "memory");
    }

    const int* srow = seq + (size_t)(batchbase + lm) * (size_t)T;
    int pp = srow[0];
    int p  = srow[1];

    float ll[8];
    const float LL0 = -2.f * logf(6.f);
    #pragma unroll
    for (int r = 0; r < 8; ++r) ll[r] = LL0;

    #pragma unroll 1
    for (int t = 2; t < T; ++t) {
        const int c = srow[t];
        if (t + 256 < T) __builtin_prefetch(srow + t + 256, 0, 0);
        const int ctx = ((pp << 2) + p) * 4 + c;   // pp*16 + p*4 + c
        pp = p; p = c;

        v8f d0 = {}; v8f d1 = {};
        matvec(d0, d1);

        #pragma unroll
        for (int r = 0; r < 8; ++r) {
            // broadcast this D-row's context from the lane that owns batch row m
            const int cx = __builtin_amdgcn_ds_bpermute((r + 8 * hl) << 2, ctx);
            const float e0 = btabS[lm * BROW + cx];
            const float e1 = btabS[(16 + lm) * BROW + cx];
            float a0 = d0[r] * e0;
            float a1 = d1[r] * e1;
            // z = sum over the 24 states (padding contributes exact zeros)
            float v = a0 + a1;
            v += __shfl_xor(v, 1);
            v += __shfl_xor(v, 2);
            v += __shfl_xor(v, 4);
            v += __shfl_xor(v, 8);
            v += 1e-30f;
            const float rz = 1.f / v;
            ll[r] += logf(v);
            const int m = r + 8 * hl;
            aw[m * AROW + lm]      = a0 * rz;
            aw[m * AROW + 16 + lm] = a1 * rz;
        }
        asm volatile("s_wait_dscnt 0x0" ::: "memory");
    }

    if (lm == 0) {
        #pragma unroll
        for (int r = 0; r < 8; ++r)
            out[batchbase + r + 8 * hl] = ll[r];
    }
}

extern "C" void kernel_launch(void* const* d_in, const int* in_sizes, int n_in,
                              void* d_out, int out_size, void* d_ws, size_t ws_size,
                              hipStream_t stream) {
    (void)n_in; (void)out_size; (void)ws_size;
    const float* w  = (const float*)d_in[0];   // transition_kernel (240)
    const float* ew = (const float*)d_in[1];   // emission_kernel (1088)
    const float* ik = (const float*)d_in[2];   // init_kernel (24)
    const int*  seq = (const int*)d_in[3];     // (2048, T) int32
    float* out = (float*)d_out;
    float* ws  = (float*)d_ws;

    const int T = in_sizes[3] / BATCH;

    prep_kernel<<<1, 256, 0, stream>>>(w, ew, ik, ws);
    const int blocks = BATCH / (16 * WPB);     // 32 blocks x 4 waves = 128 waves
    hmm_forward_kernel<<<blocks, WPB * 32, 0, stream>>>(seq, ws, out, T);
}